// Cross_Attention_3951369912700
// MI455X (gfx1250) — compile-verified
//
#include <hip/hip_runtime.h>
#include <hip/hip_bf16.h>

// ---------------------------------------------------------------------------
// Cross (channel) attention, fp32, CDNA5 gfx1250.
//   kv  : [8,192,128,128] f32      q      : [8,192,128,128] f32
//   w1  : [384,192,1,1]           w2(dw) : [384,1,3,3]
//   proj: [192,192,1,1]           scale  : [8,1,1]
//   out : [8,192,128,128] f32
// Pipeline:
//   1) kv1 = w1 @ kv           (WMMA f32 16x16x4 GEMM, per batch 384x192x16384)
//   2) kv2 = depthwise3x3(kv1)
//   3) rq/rk = 1/max(||row||2, eps)  per (b, channel)
//   4) attn[b,h,24,24] = softmax((q̂ k̂ᵀ) * scale[h])   (LDS-tiled)
//   5) Wc[b] = proj @ blockdiag(attn[b])   (192x192, tiny)
//   6) out = Wc[b] @ v         (WMMA f32 GEMM, v = kv2 channels 192..383)
// Workspace: 2*8*384*16384 + 2*8*192 + 8*8*24*24 + 8*192*192 floats (~404 MB)
// ---------------------------------------------------------------------------

typedef __attribute__((ext_vector_type(2))) float v2f;
typedef __attribute__((ext_vector_type(8))) float v8f;

#define HW   16384
#define CDIM 192

// ---------------------------------------------------------------------------
// WMMA f32 GEMM:  C[b] = A[b] (M x 192, row-major) * B[b] (192 x HW)
// One wave -> 32(M) x 64(N) tile (2 M-subtiles x 4 N-subtiles, 8 v8f accums).
// Block = 8 waves = 512 N per block.  Per K-step: 2 A-loads (b64) + 8 B-loads
// (b32) feed 8 v_wmma -> 1.25 loads per WMMA; each B frag reused by 2 WMMAs.
// Fragment layouts per CDNA5 ISA 7.12.2 (16x16x4 f32):
//   A: lane m = lane%16, a[j] = A[m][2*(lane/16)+j]
//   B: lane n = lane%16, b[j] = B[2*(lane/16)+j][n]
//   C/D: row = r + 8*(lane/16), col = lane%16
// ---------------------------------------------------------------------------
__global__ __launch_bounds__(256) void gemm_f32_wmma(
    const float* __restrict__ A, long aBatchStride,
    const float* __restrict__ B, long bBatchStride,
    float* __restrict__ C, long cBatchStride)
{
    const int wave = threadIdx.x >> 5;
    const int lane = threadIdx.x & 31;
    const int half = lane >> 4;     // 0 | 1
    const int lm   = lane & 15;
    const int klo  = half * 2;

    const int n0 = blockIdx.x * 512 + wave * 64;
    const int mt = blockIdx.y;      // 32-row M tile
    const int b  = blockIdx.z;

    const float* Ab0 = A + (long)b * aBatchStride + (long)(mt * 32 + lm) * CDIM;
    const float* Ab1 = Ab0 + 16 * CDIM;
    const float* Bb  = B + (long)b * bBatchStride;
    float*       Cb  = C + (long)b * cBatchStride;

    v8f acc0a = {}, acc1a = {}, acc2a = {}, acc3a = {};
    v8f acc0b = {}, acc1b = {}, acc2b = {}, acc3b = {};

    for (int kk = 0; kk < CDIM; kk += 4) {
        v2f a0, a1;
        a0.x = Ab0[kk + klo + 0];
        a0.y = Ab0[kk + klo + 1];
        a1.x = Ab1[kk + klo + 0];
        a1.y = Ab1[kk + klo + 1];

        const float* B0 = Bb + (long)(kk + klo) * HW + n0 + lm;  // row kk+klo
        const float* B1 = B0 + HW;                               // row kk+klo+1
        v2f b0; b0.x = B0[0];  b0.y = B1[0];
        v2f b1; b1.x = B0[16]; b1.y = B1[16];
        v2f b2; b2.x = B0[32]; b2.y = B1[32];
        v2f b3; b3.x = B0[48]; b3.y = B1[48];

        acc0a = __builtin_amdgcn_wmma_f32_16x16x4_f32(false, a0, false, b0, (short)0, acc0a, false, false);
        acc1a = __builtin_amdgcn_wmma_f32_16x16x4_f32(false, a0, false, b1, (short)0, acc1a, false, false);
        acc2a = __builtin_amdgcn_wmma_f32_16x16x4_f32(false, a0, false, b2, (short)0, acc2a, false, false);
        acc3a = __builtin_amdgcn_wmma_f32_16x16x4_f32(false, a0, false, b3, (short)0, acc3a, false, false);
        acc0b = __builtin_amdgcn_wmma_f32_16x16x4_f32(false, a1, false, b0, (short)0, acc0b, false, false);
        acc1b = __builtin_amdgcn_wmma_f32_16x16x4_f32(false, a1, false, b1, (short)0, acc1b, false, false);
        acc2b = __builtin_amdgcn_wmma_f32_16x16x4_f32(false, a1, false, b2, (short)0, acc2b, false, false);
        acc3b = __builtin_amdgcn_wmma_f32_16x16x4_f32(false, a1, false, b3, (short)0, acc3b, false, false);
    }

    float* Crow0 = Cb + (long)(mt * 32 + half * 8) * HW + n0 + lm;
    float* Crow1 = Crow0 + (long)16 * HW;
#pragma unroll
    for (int r = 0; r < 8; ++r) {
        Crow0[0]  = acc0a[r];
        Crow0[16] = acc1a[r];
        Crow0[32] = acc2a[r];
        Crow0[48] = acc3a[r];
        Crow1[0]  = acc0b[r];
        Crow1[16] = acc1b[r];
        Crow1[32] = acc2b[r];
        Crow1[48] = acc3b[r];
        Crow0 += HW;
        Crow1 += HW;
    }
}

// ---------------------------------------------------------------------------
// Depthwise 3x3, padding=1.  X,Y: [8*384][128][128], W: [384][9]
// ---------------------------------------------------------------------------
__global__ __launch_bounds__(256) void dwconv3x3(
    const float* __restrict__ X, const float* __restrict__ W,
    float* __restrict__ Y)
{
    long idx = (long)blockIdx.x * 256 + threadIdx.x;   // < 8*384*16384
    int  x  = (int)(idx & 127);
    int  y  = (int)((idx >> 7) & 127);
    long cb = idx >> 14;                 // b*384 + c
    int  c  = (int)(cb % 384);

    const float* Xc = X + (cb << 14);
    const float* w  = W + c * 9;

    float s = 0.f;
#pragma unroll
    for (int dy = -1; dy <= 1; ++dy) {
        int yy = y + dy;
        if (yy < 0 || yy > 127) continue;
#pragma unroll
        for (int dx = -1; dx <= 1; ++dx) {
            int xx = x + dx;
            if (xx < 0 || xx > 127) continue;
            s += w[(dy + 1) * 3 + (dx + 1)] * Xc[yy * 128 + xx];
        }
    }
    Y[idx] = s;
}

// ---------------------------------------------------------------------------
// Per-channel 1/max(||row||2, eps) for q and k.  Blocks [0,1536): q; rest: k.
// ---------------------------------------------------------------------------
__global__ __launch_bounds__(256) void rnorm_kernel(
    const float* __restrict__ Q, const float* __restrict__ K2,
    float* __restrict__ RQ, float* __restrict__ RK)
{
    const int id  = blockIdx.x;
    const bool isK = id >= 8 * 192;
    const int bc  = isK ? id - 8 * 192 : id;     // b*192 + c
    const int b   = bc / 192, c = bc % 192;

    const float* src = isK ? (K2 + ((long)b * 384 + c) * HW)
                           : (Q + (long)bc * HW);
    float s = 0.f;
    for (int i = threadIdx.x; i < HW; i += 256) { float v = src[i]; s += v * v; }

    __shared__ float red[256];
    red[threadIdx.x] = s;
    __syncthreads();
    for (int off = 128; off > 0; off >>= 1) {
        if (threadIdx.x < off) red[threadIdx.x] += red[threadIdx.x + off];
        __syncthreads();
    }
    if (threadIdx.x == 0) {
        float r = 1.0f / fmaxf(sqrtf(red[0]), 1e-12f);
        (isK ? RK : RQ)[bc] = r;
    }
}

// ---------------------------------------------------------------------------
// attn[b,h] = softmax( (q̂ k̂ᵀ) * scale[h] ), 24x24 per (b,head).
// One block per (b,head), 576 threads: thread t -> (c = t/24, d = t%24).
// LDS tiles 24 x 128 (rows padded to 129 to avoid bank conflicts).
// ---------------------------------------------------------------------------
#define ATNT 128
__global__ __launch_bounds__(576) void attn_kernel(
    const float* __restrict__ Q, const float* __restrict__ K2,
    const float* __restrict__ RQ, const float* __restrict__ RK,
    const float* __restrict__ scale, float* __restrict__ ATT)
{
    const int bh = blockIdx.x;          // b*8 + h
    const int b  = bh >> 3, h = bh & 7;
    const int t  = threadIdx.x;
    const int c  = t / 24, d = t % 24;

    __shared__ float qs[24][ATNT + 1];
    __shared__ float ks[24][ATNT + 1];
    __shared__ float S[24][25];

    const float* Qb = Q  + ((long)b * 192 + h * 24) * HW;
    const float* Kb = K2 + ((long)b * 384 + h * 24) * HW;
    const float* rqb = RQ + b * 192 + h * 24;
    const float* rkb = RK + b * 192 + h * 24;

    float acc = 0.f;
    for (int n0 = 0; n0 < HW; n0 += ATNT) {
        for (int j = t; j < 24 * ATNT; j += 576) {
            int row = j / ATNT, col = j % ATNT;
            qs[row][col] = Qb[(long)row * HW + n0 + col] * rqb[row];
            ks[row][col] = Kb[(long)row * HW + n0 + col] * rkb[row];
        }
        __syncthreads();
#pragma unroll 8
        for (int i = 0; i < ATNT; ++i)
            acc += qs[c][i] * ks[d][i];
        __syncthreads();
    }

    S[c][d] = acc * scale[h];
    __syncthreads();

    if (t < 24) {               // thread t owns row t's softmax
        float m = S[t][0];
#pragma unroll
        for (int j = 1; j < 24; ++j) m = fmaxf(m, S[t][j]);
        float sum = 0.f;
        float e[24];
#pragma unroll
        for (int j = 0; j < 24; ++j) { e[j] = __expf(S[t][j] - m); sum += e[j]; }
        float inv = 1.0f / sum;
        float* dst = ATT + ((long)bh * 24 + t) * 24;
#pragma unroll
        for (int j = 0; j < 24; ++j) dst[j] = e[j] * inv;
    }
}

// ---------------------------------------------------------------------------
// Wc[b][e][g] = sum_i proj[e][head(g)*24+i] * attn[b][head(g)][i][g%24]
// ---------------------------------------------------------------------------
__global__ __launch_bounds__(256) void wc_kernel(
    const float* __restrict__ P, const float* __restrict__ ATT,
    float* __restrict__ WC)
{
    int idx = blockIdx.x * 256 + threadIdx.x;   // < 8*192*192
    int g = idx % 192;
    int e = (idx / 192) % 192;
    int b = idx / (192 * 192);
    int h = g / 24, gj = g % 24;

    const float* Prow = P + e * 192 + h * 24;
    const float* Acol = ATT + ((long)(b * 8 + h) * 24) * 24 + gj;  // stride 24 over i
    float s = 0.f;
#pragma unroll
    for (int i = 0; i < 24; ++i) s += Prow[i] * Acol[i * 24];
    WC[idx] = s;
}

// ---------------------------------------------------------------------------
extern "C" void kernel_launch(void* const* d_in, const int* in_sizes, int n_in,
                              void* d_out, int out_size, void* d_ws, size_t ws_size,
                              hipStream_t stream)
{
    const float* kv    = (const float*)d_in[0];  // [8,192,128,128]
    const float* q     = (const float*)d_in[1];  // [8,192,128,128]
    const float* w1    = (const float*)d_in[2];  // [384,192]
    const float* w2    = (const float*)d_in[3];  // [384,9]
    const float* projw = (const float*)d_in[4];  // [192,192]
    const float* scale = (const float*)d_in[5];  // [8]
    float*       out   = (float*)d_out;

    float* ws  = (float*)d_ws;
    float* kv1 = ws;                                  // 8*384*HW
    float* kv2 = kv1 + (size_t)8 * 384 * HW;          // 8*384*HW
    float* rq  = kv2 + (size_t)8 * 384 * HW;          // 8*192
    float* rk  = rq + 8 * 192;                        // 8*192
    float* att = rk + 8 * 192;                        // 8*8*24*24
    float* wc  = att + 8 * 8 * 24 * 24;               // 8*192*192

    // 1) kv1 = w1 @ kv   (M=384 -> 12 M-tiles of 32, shared A across batches)
    gemm_f32_wmma<<<dim3(32, 12, 8), 256, 0, stream>>>(
        w1, 0L, kv, (long)192 * HW, kv1, (long)384 * HW);

    // 2) kv2 = dwconv3x3(kv1)
    dwconv3x3<<<(8 * 384 * HW) / 256, 256, 0, stream>>>(kv1, w2, kv2);

    // 3) reciprocal L2 norms of q rows and k rows (k = kv2 channels 0..191)
    rnorm_kernel<<<2 * 8 * 192, 256, 0, stream>>>(q, kv2, rq, rk);

    // 4) attention matrices + softmax
    attn_kernel<<<64, 576, 0, stream>>>(q, kv2, rq, rk, scale, att);

    // 5) combined per-batch weight Wc = proj @ blockdiag(attn)
    wc_kernel<<<(8 * 192 * 192) / 256, 256, 0, stream>>>(projw, att, wc);

    // 6) out = Wc[b] @ v   (v = kv2 channels 192..383, M=192 -> 6 M-tiles)
    gemm_f32_wmma<<<dim3(32, 6, 8), 256, 0, stream>>>(
        wc, (long)192 * 192, kv2 + (size_t)192 * HW, (long)384 * HW,
        out, (long)192 * HW);
}